// STnet_52183852646437
// MI455X (gfx1250) — compile-verified
//
#include <hip/hip_runtime.h>

typedef __attribute__((ext_vector_type(2))) float v2f;
typedef __attribute__((ext_vector_type(8))) float v8f;

// ---------------------------------------------------------------- utilities
__global__ void zero_f32(float* __restrict__ p, int n) {
    int i = blockIdx.x * blockDim.x + threadIdx.x;
    if (i < n) p[i] = 0.0f;
}

// ---------------------------------------------------------------- degree
__global__ void deg_init(float* __restrict__ deg, int n) {
    int i = blockIdx.x * blockDim.x + threadIdx.x;
    if (i < n) deg[i] = 1.0f;                       // self loop contributes 1
}
__global__ void deg_accum(const int* __restrict__ dst, float* __restrict__ deg, int E) {
    int e = blockIdx.x * blockDim.x + threadIdx.x;
    if (e < E) atomicAdd(&deg[dst[e]], 1.0f);
}
__global__ void deg_rsqrt(float* __restrict__ deg, int n) {
    int i = blockIdx.x * blockDim.x + threadIdx.x;
    if (i < n) deg[i] = rsqrtf(deg[i]);             // deg >= 1
}

// ---------------------------------------------------------------- fp32 WMMA GEMM
// Y[M,64] = act(X[M,64]) @ W[64,64] + bias ; ACT => relu(x*scale + shift) on A.
// 128 threads = 4 waves; wave w computes rows [blk*64 + w*16, +16) x all 64 cols.
// ACT is a template parameter so the K-loop is branch-free straight-line code.
template <bool ACT>
__global__ __launch_bounds__(128)
void gemm64_wmma(const float* __restrict__ X, const float* __restrict__ W,
                 const float* __restrict__ bias,
                 const float* __restrict__ bnscale, const float* __restrict__ bnshift,
                 float* __restrict__ Y, int M)
{
    __shared__ float sW[64 * 64];
    __shared__ float sScale[64], sShift[64];
    const int tid = threadIdx.x;
    for (int i = tid; i < 64 * 64; i += 128) sW[i] = W[i];
    if (ACT && tid < 64) { sScale[tid] = bnscale[tid]; sShift[tid] = bnshift[tid]; }
    __syncthreads();

    const int wave = tid >> 5;
    const int lane = tid & 31;
    const int row0 = blockIdx.x * 64 + wave * 16;
    const int nlo  = lane & 15;              // M index for A loads, N index for B/C/D
    const int khi  = (lane >> 4) << 1;       // 0 for lanes 0-15, 2 for lanes 16-31

    int r = row0 + nlo; if (r > M - 1) r = M - 1;   // clamp: keep EXEC all-1s for WMMA
    const float* xrow = X + (size_t)r * 64;

    v8f acc[4];
    for (int ct = 0; ct < 4; ++ct) {
        float bv = bias[ct * 16 + nlo];      // bias depends on column only
        #pragma unroll
        for (int g = 0; g < 8; ++g) acc[ct][g] = bv;
    }

    #pragma unroll
    for (int k0 = 0; k0 < 64; k0 += 4) {
        const int ka = k0 + khi;
        v2f a;
        a.x = xrow[ka];
        a.y = xrow[ka + 1];
        if (ACT) {
            a.x = fmaxf(a.x * sScale[ka]     + sShift[ka],     0.0f);
            a.y = fmaxf(a.y * sScale[ka + 1] + sShift[ka + 1], 0.0f);
        }
        #pragma unroll
        for (int ct = 0; ct < 4; ++ct) {
            v2f b;
            b.x = sW[ka * 64       + ct * 16 + nlo];
            b.y = sW[(ka + 1) * 64 + ct * 16 + nlo];
            acc[ct] = __builtin_amdgcn_wmma_f32_16x16x4_f32(
                false, a, false, b, (short)0, acc[ct], false, false);
        }
    }

    const int mhi = (lane >> 4) * 8;
    for (int ct = 0; ct < 4; ++ct) {
        #pragma unroll
        for (int g = 0; g < 8; ++g) {
            int m = row0 + mhi + g;
            if (m < M) Y[(size_t)m * 64 + ct * 16 + nlo] = acc[ct][g];
        }
    }
}

// ---------------------------------------------------------------- propagation
// out[i] = t[i]*dis[i]^2 (self loop), then out[dst] += t[src]*dis[src]*dis[dst]
__global__ void self_loop_init(const float* __restrict__ t, const float* __restrict__ dis,
                               float* __restrict__ out, int n)
{
    int gid = blockIdx.x * blockDim.x + threadIdx.x;
    if (gid >= n * 64) return;
    int i = gid >> 6;
    float d = dis[i];
    out[gid] = t[gid] * d * d;
}
__global__ void scatter_edges(const float* __restrict__ t, const float* __restrict__ dis,
                              const int* __restrict__ src, const int* __restrict__ dst,
                              float* __restrict__ out, int E)
{
    int gid = blockIdx.x * blockDim.x + threadIdx.x;
    int e = gid >> 6;
    if (e >= E) return;
    int f = gid & 63;
    int s = src[e], d = dst[e];
    float nrm = dis[s] * dis[d];
    atomicAdd(&out[(size_t)d * 64 + f], t[(size_t)s * 64 + f] * nrm);
}

// ---------------------------------------------------------------- batch norm
__global__ __launch_bounds__(256)
void bn_stats(const float* __restrict__ h, int n, float* __restrict__ stats)
{
    const int tid = threadIdx.x;
    const int f = tid & 63;
    const int g = tid >> 6;                  // 4 row groups per block
    float s = 0.f, q = 0.f;
    for (int r = blockIdx.x * 4 + g; r < n; r += gridDim.x * 4) {
        float v = h[(size_t)r * 64 + f];
        s += v; q += v * v;
    }
    __shared__ float ls[256], lq[256];
    ls[tid] = s; lq[tid] = q;
    __syncthreads();
    if (tid < 64) {
        s = ls[tid] + ls[tid + 64] + ls[tid + 128] + ls[tid + 192];
        q = lq[tid] + lq[tid + 64] + lq[tid + 128] + lq[tid + 192];
        atomicAdd(&stats[tid], s);
        atomicAdd(&stats[64 + tid], q);
    }
}
__global__ void bn_finalize(const float* __restrict__ stats,
                            const float* __restrict__ gamma, const float* __restrict__ beta,
                            float invn, float* __restrict__ scale, float* __restrict__ shift)
{
    int f = threadIdx.x;
    if (f >= 64) return;
    float mean = stats[f] * invn;
    float var  = stats[64 + f] * invn - mean * mean;
    float sc = gamma[f] * rsqrtf(var + 1e-5f);
    scale[f] = sc;
    shift[f] = beta[f] - mean * sc;
}

// ---------------------------------------------------------------- graph pooling
__global__ void pool_accum(const float* __restrict__ emb, const int* __restrict__ batch,
                           float* __restrict__ sums, float* __restrict__ counts, int n)
{
    int gid = blockIdx.x * blockDim.x + threadIdx.x;
    if (gid >= n * 64) return;
    int i = gid >> 6, f = gid & 63;
    int g = batch[i];
    atomicAdd(&sums[(size_t)g * 64 + f], emb[gid]);
    if (f == 0) atomicAdd(&counts[g], 1.0f);
}
// block per graph (64 threads): mean, L2 normalize, write rep
__global__ __launch_bounds__(64)
void pool_finalize(const float* __restrict__ sums, const float* __restrict__ counts,
                   float* __restrict__ rep)
{
    int g = blockIdx.x, f = threadIdx.x;
    float c = fmaxf(counts[g], 1.0f);
    float v = sums[(size_t)g * 64 + f] / c;
    __shared__ float sh[64];
    sh[f] = v * v;
    __syncthreads();
    for (int s = 32; s > 0; s >>= 1) {
        if (f < s) sh[f] += sh[f + s];
        __syncthreads();
    }
    rep[(size_t)g * 64 + f] = v / sqrtf(sh[0]);
}

// ---------------------------------------------------------------- classifier tail
// z2[g,c] = bc2[c] + sum_f relu(zc[g,f]*scale[f]+shift[f]) * Wc2[f,c]
__global__ void cls_final(const float* __restrict__ zc,
                          const float* __restrict__ scale, const float* __restrict__ shift,
                          const float* __restrict__ Wc2, const float* __restrict__ bc2,
                          float* __restrict__ z2, int G)
{
    int gid = blockIdx.x * blockDim.x + threadIdx.x;
    if (gid >= G * 10) return;
    int g = gid / 10, c = gid % 10;
    float acc = bc2[c];
    for (int f = 0; f < 64; ++f) {
        float v = fmaxf(zc[(size_t)g * 64 + f] * scale[f] + shift[f], 0.0f);
        acc += v * Wc2[f * 10 + c];
    }
    z2[gid] = acc;
}
__global__ void log_softmax10(const float* __restrict__ z2, float* __restrict__ pred, int G)
{
    int g = blockIdx.x * blockDim.x + threadIdx.x;
    if (g >= G) return;
    float m = -3.4e38f;
    for (int c = 0; c < 10; ++c) m = fmaxf(m, z2[g * 10 + c]);
    float s = 0.f;
    for (int c = 0; c < 10; ++c) s += expf(z2[g * 10 + c] - m);
    float l = m + logf(s);
    for (int c = 0; c < 10; ++c) pred[g * 10 + c] = z2[g * 10 + c] - l;
}

// ---------------------------------------------------------------- driver
extern "C" void kernel_launch(void* const* d_in, const int* in_sizes, int n_in,
                              void* d_out, int out_size, void* d_ws, size_t ws_size,
                              hipStream_t stream)
{
    const float* x      = (const float*)d_in[0];
    const int*   eidx   = (const int*)  d_in[1];   // [2,E] row-major: src row 0, dst row 1
    const int*   batch  = (const int*)  d_in[2];
    const float* W1     = (const float*)d_in[3];
    const float* b1     = (const float*)d_in[4];
    const float* gamma1 = (const float*)d_in[5];
    const float* beta1  = (const float*)d_in[6];
    const float* W2     = (const float*)d_in[7];
    const float* b2     = (const float*)d_in[8];
    const float* Wc1    = (const float*)d_in[9];
    const float* bc1    = (const float*)d_in[10];
    const float* gammac = (const float*)d_in[11];
    const float* betac  = (const float*)d_in[12];
    const float* Wc2    = (const float*)d_in[13];
    const float* bc2    = (const float*)d_in[14];

    const int N = in_sizes[0] / 64;
    const int E = in_sizes[1] / 2;
    const int G = out_size / 74;              // out = G*10 (pred) + G*64 (rep)
    const int* src = eidx;
    const int* dst = eidx + E;

    float* ws = (float*)d_ws;
    size_t off = 0;
    float* A      = ws + off; off += (size_t)N * 64;
    float* B      = ws + off; off += (size_t)N * 64;
    float* dis    = ws + off; off += (size_t)N;
    float* stats  = ws + off; off += 128;
    float* scale  = ws + off; off += 64;
    float* shift  = ws + off; off += 64;
    float* sums   = ws + off; off += (size_t)G * 64;
    float* counts = ws + off; off += (size_t)G;
    float* zc     = ws + off; off += (size_t)G * 64;
    float* z2     = ws + off; off += (size_t)G * 10;

    float* outf = (float*)d_out;
    float* rep  = outf + (size_t)G * 10;      // graph_rep region of the output

    const int T = 256;
    const int nfBlocks = (int)(((size_t)N * 64 + T - 1) / T);
    const int efBlocks = (int)(((size_t)E * 64 + T - 1) / T);

    // degrees -> dis = rsqrt(deg)
    deg_init  <<<(N + T - 1) / T, T, 0, stream>>>(dis, N);
    deg_accum <<<(E + T - 1) / T, T, 0, stream>>>(dst, dis, E);
    deg_rsqrt <<<(N + T - 1) / T, T, 0, stream>>>(dis, N);

    // conv1: A = x@W1 + b1 ; B = propagate(A)
    gemm64_wmma<false><<<(N + 63) / 64, 128, 0, stream>>>(x, W1, b1, nullptr, nullptr, A, N);
    self_loop_init<<<nfBlocks, T, 0, stream>>>(A, dis, B, N);
    scatter_edges <<<efBlocks, T, 0, stream>>>(A, dis, src, dst, B, E);

    // BN1 stats; conv2 GEMM fuses relu(BN(B)) into its A-load: A = relu(BN(B))@W2 + b2
    zero_f32<<<1, 128, 0, stream>>>(stats, 128);
    bn_stats<<<512, 256, 0, stream>>>(B, N, stats);
    bn_finalize<<<1, 64, 0, stream>>>(stats, gamma1, beta1, 1.0f / (float)N, scale, shift);
    gemm64_wmma<true><<<(N + 63) / 64, 128, 0, stream>>>(B, W2, b2, scale, shift, A, N);

    // node_emb: B = propagate(A)
    self_loop_init<<<nfBlocks, T, 0, stream>>>(A, dis, B, N);
    scatter_edges <<<efBlocks, T, 0, stream>>>(A, dis, src, dst, B, E);

    // per-graph mean pool + L2 normalize -> rep (directly into output)
    zero_f32<<<(G * 65 + T - 1) / T, T, 0, stream>>>(sums, G * 65);  // sums + counts contiguous
    pool_accum<<<nfBlocks, T, 0, stream>>>(B, batch, sums, counts, N);
    pool_finalize<<<G, 64, 0, stream>>>(sums, counts, rep);

    // classifier: zc = rep@Wc1 + bc1 ; BN over G rows ; relu + @Wc2 + bc2 ; log_softmax
    gemm64_wmma<false><<<(G + 63) / 64, 128, 0, stream>>>(rep, Wc1, bc1, nullptr, nullptr, zc, G);
    zero_f32<<<1, 128, 0, stream>>>(stats, 128);
    bn_stats<<<64, 256, 0, stream>>>(zc, G, stats);
    bn_finalize<<<1, 64, 0, stream>>>(stats, gammac, betac, 1.0f / (float)G, scale, shift);
    cls_final<<<(G * 10 + T - 1) / T, T, 0, stream>>>(zc, scale, shift, Wc2, bc2, z2, G);
    log_softmax10<<<(G + T - 1) / T, T, 0, stream>>>(z2, outf, G);

    (void)n_in; (void)ws_size;
}